// RelationalFeatures_54623394070800
// MI455X (gfx1250) — compile-verified
//
#include <hip/hip_runtime.h>
#include <math.h>

typedef __attribute__((ext_vector_type(2))) float v2f;
typedef __attribute__((ext_vector_type(8))) float v8f;

#define GROUPS_TOTAL 65536      // 32 batches * (65536 pairs / 32 per group)
#define WAVES_PER_BLOCK 4
#define HB_STRIDE 66            // h buffer row stride (pad vs 64 banks)
#define OB_STRIDE 17            // out-stage row stride (odd -> conflict free)
#define FB_STRIDE 10            // feature row stride

__device__ __forceinline__ void wait_ds0() {
  asm volatile("s_wait_dscnt 0" ::: "memory");
}

__device__ __forceinline__ v8f wmma_f32(v2f a, v2f b, v8f c) {
  // D = A(16x4 f32) * B(4x16 f32) + C(16x16 f32)
  return __builtin_amdgcn_wmma_f32_16x16x4_f32(
      /*neg_a=*/false, a, /*neg_b=*/false, b,
      /*c_mod=*/(short)0, c, /*reuse_a=*/false, /*reuse_b=*/false);
}

// fast hardware approximations (v_rcp_f32 / v_rsq_f32, ~1 ulp)
__device__ __forceinline__ float fast_rcp(float x) { return __builtin_amdgcn_rcpf(x); }
__device__ __forceinline__ float fast_rsq(float x) { return __builtin_amdgcn_rsqf(x); }

__global__ __launch_bounds__(128) void relfeat_mlp_kernel(
    const float* __restrict__ pos, const float* __restrict__ vel,
    const float* __restrict__ att, const float* __restrict__ W1,
    const float* __restrict__ b1, const float* __restrict__ W2,
    const float* __restrict__ b2, float* __restrict__ out, int totalWaves) {
  __shared__ float lds[WAVES_PER_BLOCK * (32 * HB_STRIDE + 32 * OB_STRIDE)];
  const int lane = threadIdx.x & 31;
  const int wave = threadIdx.x >> 5;
  float* hbuf = lds + wave * (32 * HB_STRIDE + 32 * OB_STRIDE);
  float* obuf = hbuf + 32 * HB_STRIDE;
  float* fbuf = obuf;  // aliased: features and out-stage used in disjoint phases

  const int m16 = lane & 15;  // row / column within 16-tile
  const int hi  = lane >> 4;  // half-wave select (K offset of 2 in frags)

  // ---- per-wave constant weight fragments (B layout: lanes=N, VGPRs=K, upper half K+2)
  v2f w1b[4][2];
#pragma unroll
  for (int t = 0; t < 4; ++t) {
#pragma unroll
    for (int f = 0; f < 2; ++f) {
      int row = f * 4 + hi * 2;
      int col = t * 16 + m16;
      w1b[t][f].x = W1[row * 64 + col];
      w1b[t][f].y = W1[(row + 1) * 64 + col];
    }
  }
  v2f w2b[16];
#pragma unroll
  for (int k = 0; k < 16; ++k) {
    int row = k * 4 + hi * 2;
    w2b[k].x = W2[row * 16 + m16];
    w2b[k].y = W2[(row + 1) * 16 + m16];
  }
  float b1v[4];
#pragma unroll
  for (int t = 0; t < 4; ++t) b1v[t] = b1[t * 16 + m16];
  const float b2v = b2[m16];

  const int waveGlobal = blockIdx.x * WAVES_PER_BLOCK + wave;
  for (int g = waveGlobal; g < GROUPS_TOTAL; g += totalWaves) {
    const int b  = g >> 11;            // batch (2048 groups per batch)
    const int p0 = (g & 2047) << 5;    // first pair index within batch
    const int p  = p0 + lane;          // this lane's pair
    const int i  = p >> 8;
    const int j  = p & 255;

    // ---- per-lane feature computation (1 pair per lane) ----
    const float* pb = pos + b * 512;
    const float* vb = vel + b * 512;
    const float* ab = att + b * 512;
    float pix = pb[2 * i + 0], piy = pb[2 * i + 1];
    float pjx = pb[2 * j + 0], pjy = pb[2 * j + 1];
    float vix = vb[2 * i + 0], viy = vb[2 * i + 1];
    float vjx = vb[2 * j + 0], vjy = vb[2 * j + 1];
    float aix = ab[2 * i + 0], aiy = ab[2 * i + 1];
    float ajx = ab[2 * j + 0], ajy = ab[2 * j + 1];

    float rpx = pjx - pix, rpy = pjy - piy;   // rel_pos[i][j] = pos[j]-pos[i]
    float rvx = vjx - vix, rvy = vjy - viy;
    float d2 = fmaxf(rpx * rpx + rpy * rpy, 1e-12f);
    float rinv = fast_rsq(d2);                 // 1/sqrt(d2), single v_rsq_f32
    float dist = d2 * rinv;                    // sqrt(d2)
    float dx = rpx * rinv, dy = rpy * rinv;
    float bear_cos = aix * dx + aiy * dy;
    float bear_sin = aix * dy - aiy * dx;
    float asp_cos = -(ajx * dx + ajy * dy);
    float asp_sin = -(ajx * dy - ajy * dx);
    float symlog_dist = __logf(dist + 1.0f);
    float closing = -(rvx * dx + rvy * dy);
    float symlog_close = copysignf(__logf(fabsf(closing) + 1.0f), closing);
    float rel_head_cos = aix * ajx + aiy * ajy;
    float rel_head_sin = aix * ajy - aiy * ajx;

    float* fr = fbuf + lane * FB_STRIDE;
    fr[0] = bear_sin;    fr[1] = bear_cos;
    fr[2] = asp_sin;     fr[3] = asp_cos;
    fr[4] = symlog_dist; fr[5] = symlog_close;
    fr[6] = rel_head_sin; fr[7] = rel_head_cos;
    wait_ds0();

    // ---- layer 1 (8 -> 64) + SiLU, two 16-pair M-tiles ----
#pragma unroll
    for (int tau = 0; tau < 2; ++tau) {
      const float* far = fbuf + (tau * 16 + m16) * FB_STRIDE + hi * 2;
      v2f a0; a0.x = far[0]; a0.y = far[1];   // K = 0..3
      v2f a1; a1.x = far[4]; a1.y = far[5];   // K = 4..7
#pragma unroll
      for (int t = 0; t < 4; ++t) {
        v8f c;
#pragma unroll
        for (int e = 0; e < 8; ++e) c[e] = b1v[t];
        c = wmma_f32(a0, w1b[t][0], c);
        c = wmma_f32(a1, w1b[t][1], c);
#pragma unroll
        for (int v = 0; v < 8; ++v) {
          float x = c[v];
          // SiLU: x * rcp(1 + exp(-x)) -> v_exp_f32 + v_rcp_f32 + muls, no div chain
          float s = x * fast_rcp(1.0f + __expf(-x));
          hbuf[(tau * 16 + v + hi * 8) * HB_STRIDE + t * 16 + m16] = s;
        }
      }
    }
    wait_ds0();

    // ---- layer 2 (64 -> 16) ----
#pragma unroll
    for (int tau = 0; tau < 2; ++tau) {
      v8f c2;
#pragma unroll
      for (int e = 0; e < 8; ++e) c2[e] = b2v;
      const float* hrow = hbuf + (tau * 16 + m16) * HB_STRIDE + hi * 2;
#pragma unroll
      for (int k = 0; k < 16; ++k) {
        v2f a; a.x = hrow[k * 4 + 0]; a.y = hrow[k * 4 + 1];
        c2 = wmma_f32(a, w2b[k], c2);
      }
#pragma unroll
      for (int v = 0; v < 8; ++v)
        obuf[(tau * 16 + v + hi * 8) * OB_STRIDE + m16] = c2[v];
    }
    wait_ds0();

    // ---- transpose-stage write-out: 128B-contiguous stores per channel ----
    const float* orow = obuf + lane * OB_STRIDE;
    const long obase = ((long)(b * 16) << 16) + p0 + lane;
#pragma unroll
    for (int r = 0; r < 16; ++r)
      out[obase + ((long)r << 16)] = orow[r];
    wait_ds0();  // keep obuf reads ahead of next iteration's fbuf alias writes
  }
}

extern "C" void kernel_launch(void* const* d_in, const int* in_sizes, int n_in,
                              void* d_out, int out_size, void* d_ws, size_t ws_size,
                              hipStream_t stream) {
  const float* pos = (const float*)d_in[0];
  const float* vel = (const float*)d_in[1];
  const float* att = (const float*)d_in[2];
  const float* W1  = (const float*)d_in[3];
  const float* b1  = (const float*)d_in[4];
  const float* W2  = (const float*)d_in[5];
  const float* b2  = (const float*)d_in[6];
  float* out = (float*)d_out;

  const int blocks = 4096;                      // 16384 waves, 4 groups each
  const int totalWaves = blocks * WAVES_PER_BLOCK;
  relfeat_mlp_kernel<<<blocks, 128, 0, stream>>>(pos, vel, att, W1, b1, W2, b2,
                                                 out, totalWaves);
}